// ConwayLayer_63934883168794
// MI455X (gfx1250) — compile-verified
//
#include <hip/hip_runtime.h>
#include <cstdint>

// Continuous Game-of-Life, 8 steps fused via temporal blocking in LDS.
// 16 x 1024 x 1024 f32, periodic boundary.
//
// Tile: 64x64 interior + 8 halo -> 80x80 LDS buffer, ping-pong (2 x 25600 B).
// HBM traffic is at the floor: each cell read once (async b128 global->LDS DMA,
// per-lane wrapped addresses) and written once (async b128 LDS->global DMA).
// Compute: thread = (column, row-chunk); walks rows reusing horizontal sums in
// registers -> 3 LDS loads + 1 store per cell-step.

#define TILE     64
#define HALO     8
#define BUF      (TILE + 2 * HALO)     // 80
#define BUFN     (BUF * BUF)           // 6400
#define NT       320                   // 80 columns x 4 row-chunks = 10 wave32
#define ROWS_PER 20                    // 80 rows / 4 chunks
#define STEPS    8
#define IMG_W    1024
#define IMG_H    1024

__device__ __forceinline__ void async_load_b128_to_lds(uint32_t lds_addr, const float* gptr) {
    asm volatile("global_load_async_to_lds_b128 %0, %1, off"
                 :: "v"(lds_addr), "v"((uint64_t)gptr)
                 : "memory");
}

__device__ __forceinline__ void async_store_b128_from_lds(const float* gptr, uint32_t lds_addr) {
    asm volatile("global_store_async_from_lds_b128 %0, %1, off"
                 :: "v"((uint64_t)gptr), "v"(lds_addr)
                 : "memory");
}

__device__ __forceinline__ void wait_asynccnt0() {
    asm volatile("s_wait_asynccnt 0" ::: "memory");
}

__device__ __forceinline__ float gol_update(float n, float x) {
    const float d3 = n - 3.0f;
    const float d2 = n - 2.0f;
    // sigmoid(-50*d^2) = 1/(1 + exp(50*d^2)); overflow -> rcp(inf) = 0, correct
    const float e3 = __expf(50.0f * d3 * d3);
    const float e2 = __expf(50.0f * d2 * d2);
    const float s3 = __builtin_amdgcn_rcpf(1.0f + e3);
    const float s2 = __builtin_amdgcn_rcpf(1.0f + e2);
    const float x2 = x * x, x4 = x2 * x2;
    const float y  = 1.0f - x;
    const float y2 = y * y, y4 = y2 * y2;
    float o = 2.0f * x4 * (s2 + s3) + 2.0f * y4 * s3;
    return fminf(fmaxf(o, 0.0f), 1.0f);
}

__global__ __launch_bounds__(NT)
void conway8_kernel(const float* __restrict__ src, float* __restrict__ dst)
{
    __shared__ __align__(16) float bufA[BUFN];
    __shared__ __align__(16) float bufB[BUFN];

    const int tid = threadIdx.x;
    const int tx  = blockIdx.x;
    const int ty  = blockIdx.y;
    const int img = blockIdx.z;

    const float* base = src + (size_t)img * (size_t)(IMG_W * IMG_H);

    // ---- async halo-tile load: 1600 vec4 groups; wrap-safe b128 segments ----
    // (halo offset, tile size, image width all = 0 mod 4 -> a 4-float group
    //  never straddles the periodic boundary, and is 16B aligned both sides)
    for (int g = tid; g < BUFN / 4; g += NT) {
        const int r  = g / (BUF / 4);
        const int c4 = (g % (BUF / 4)) * 4;
        const int gy = (ty * TILE + r  - HALO) & (IMG_H - 1);
        const int gx = (tx * TILE + c4 - HALO) & (IMG_W - 1);
        async_load_b128_to_lds((uint32_t)(uint64_t)&bufA[r * BUF + c4],
                               base + (gy * IMG_W + gx));
    }
    wait_asynccnt0();
    __syncthreads();

    // fixed thread mapping for all 8 steps: column c, row chunk [rq0, rq1)
    const int c   = tid % BUF;        // 0..79
    const int q   = tid / BUF;        // 0..3
    const int rq0 = q * ROWS_PER;
    const int rq1 = rq0 + ROWS_PER;

    float* rd = bufA;
    float* wr = bufB;

    #pragma unroll
    for (int s = 0; s < STEPS; ++s) {
        const int m = s + 1;                       // shrinking valid margin
        if (c >= m && c < BUF - m) {
            const int r0 = (rq0 > m) ? rq0 : m;
            const int r1 = (rq1 < BUF - m) ? rq1 : (BUF - m);
            if (r0 < r1) {
                // prime H(r0-1), H(r0); keep center x in a register
                const float* row = rd + (r0 - 1) * BUF + c;
                float hm = row[-1] + row[0] + row[1];
                row += BUF;
                float xc = row[0];
                float hc = row[-1] + xc + row[1];
                float* wp = wr + r0 * BUF + c;
                for (int r = r0; r < r1; ++r) {
                    row += BUF;
                    const float xn = row[0];
                    const float hn = row[-1] + xn + row[1];
                    const float n  = hm + hc + hn - xc;   // 8-neighbor sum
                    *wp = gol_update(n, xc);
                    wp += BUF;
                    hm = hc; hc = hn; xc = xn;
                }
            }
        }
        __syncthreads();
        float* t = rd; rd = wr; wr = t;
    }
    // after 8 swaps, `rd` holds the step-8 result

    // ---- async interior writeback: 1024 vec4 groups, all 16B aligned ----
    float* obase = dst + (size_t)img * (size_t)(IMG_W * IMG_H);
    for (int g = tid; g < (TILE * TILE) / 4; g += NT) {
        const int r  = g >> 4;            // / (TILE/4)
        const int c4 = (g & 15) * 4;
        float* gp = obase + ((ty * TILE + r) * IMG_W + (tx * TILE + c4));
        async_store_b128_from_lds(gp,
            (uint32_t)(uint64_t)&rd[(r + HALO) * BUF + (c4 + HALO)]);
    }
    wait_asynccnt0();
}

extern "C" void kernel_launch(void* const* d_in, const int* in_sizes, int n_in,
                              void* d_out, int out_size, void* d_ws, size_t ws_size,
                              hipStream_t stream) {
    const float* x = (const float*)d_in[0];
    float* out = (float*)d_out;

    const int batch = in_sizes[0] / (IMG_W * IMG_H);   // 16 for reference shapes
    dim3 grid(IMG_W / TILE, IMG_H / TILE, batch);
    conway8_kernel<<<grid, NT, 0, stream>>>(x, out);
}